// Model_18305150616283
// MI455X (gfx1250) — compile-verified
//
#include <hip/hip_runtime.h>

typedef __attribute__((ext_vector_type(2))) float v2f;
typedef __attribute__((ext_vector_type(8))) float v8f;

constexpr int SEQ  = 720;
constexpr int INL  = 360;   // SEQ/2
constexpr int PRED = 336;
constexpr int CH   = 862;
constexpr int RK   = 32;
constexpr int NB   = 256;   // batch
constexpr int NT   = 64;    // channel tile per workgroup

// workspace layout (floats)
constexpr int WS_W1   = 0;                    // [32][360]  W1^T  (row r, col n)
constexpr int WS_W2   = 32 * INL;             // [336][32]  W2^T  (row j, col r)
constexpr int WS_BIAS = WS_W2 + PRED * RK;    // [336]
constexpr int WS_COEF = WS_BIAS + PRED;       // [336]

// LDS layout (dwords). H and Y use K-pair interleaved layout:
//   elem (k, c) at [(k>>1)*PITCH + 2*c + (k&1)]
// PITCH = 160 (mod 64 == 32) -> lane halves (pair p vs p+1) hit disjoint
// 32-bank halves; a wave32 ds_load_b64 covers all 64 banks once.
constexpr int HP2 = 160;                      // >= 2*NT, pad for banks
constexpr int YP2 = 160;
constexpr int W1P = 362;                      // even (b64 aligned)
constexpr int W2P = 34;                       // even (b64 aligned)

constexpr int L_H    = 0;                     // 180 * HP2 = 28800
constexpr int L_Y    = L_H + 180 * HP2;       // 16 * YP2  =  2560
constexpr int L_W2   = L_Y + 16 * YP2;        // 336 * W2P = 11424
constexpr int L_W1   = L_W2 + PRED * W2P;     // 32 * W1P  = 11584
constexpr int L_BIAS = L_W1 + 32 * W1P;       // 336
constexpr int L_COEF = L_BIAS + PRED;         // 336
constexpr int L_COL  = L_COEF + PRED;         // 64
constexpr int SMEM_DW = L_COL + NT;

// ---------------------------------------------------------------------------
// Prep A: build fused weight matrices from on-the-fly DCT/IDCT cosine tables.
//   W1^T[r][n] = (1/360) * sum_k D[k,n] * A[k,r]
//   W2^T[j][r] = sum_m B[r,m] * M[m,j]
//   bias2[j]   = sum_m b[m] * M[m,j]
// ---------------------------------------------------------------------------
__global__ void prep_a(const float* __restrict__ A, const float* __restrict__ Bm,
                       const float* __restrict__ bv, float* __restrict__ ws) {
  const int tid = blockIdx.x * blockDim.x + threadIdx.x;
  const double PI = 3.14159265358979323846;
  if (tid < 32 * INL) {
    const int r = tid / INL, n = tid % INL;
    const double s0 = sqrt(1.0 / (4.0 * INL));
    const double s1 = sqrt(1.0 / (2.0 * INL));
    float acc = 0.f;
    for (int k = 0; k < INL; ++k) {
      double d = ((k == 0) ? s0 : s1) * 2.0 * cos(PI * k * (2 * n + 1) / (2.0 * INL));
      acc += (float)d * A[k * RK + r];
    }
    ws[WS_W1 + r * INL + n] = acc / (float)INL;
  } else if (tid < 32 * INL + PRED * RK) {
    const int e = tid - 32 * INL;
    const int j = e / RK, r = e % RK;
    float acc = 0.f;
    for (int m = 0; m < PRED; ++m) {
      double v = cos(PI * m * (2 * j + 1) / (2.0 * PRED)) / PRED;
      if (m == 0) v *= 0.5;
      acc += Bm[r * PRED + m] * (float)v;
    }
    ws[WS_W2 + e] = acc;                 // e == j*RK + r
  } else if (tid < 32 * INL + PRED * RK + PRED) {
    const int j = tid - (32 * INL + PRED * RK);
    float acc = 0.f;
    for (int m = 0; m < PRED; ++m) {
      double v = cos(PI * m * (2 * j + 1) / (2.0 * PRED)) / PRED;
      if (m == 0) v *= 0.5;
      acc += bv[m] * (float)v;
    }
    ws[WS_BIAS + j] = acc;
  }
}

// Prep B: coef[j] = 1 - sqrt(2) * sum_r (sum_n W1^T[r][n]) * W2^T[j][r]
__global__ void prep_b(float* __restrict__ ws) {
  __shared__ float s[RK];
  const int tid = threadIdx.x;
  if (tid < RK) {
    float a = 0.f;
    for (int n = 0; n < INL; ++n) a += ws[WS_W1 + tid * INL + n];
    s[tid] = a;
  }
  __syncthreads();
  if (tid < PRED) {
    float t = 0.f;
    for (int r = 0; r < RK; ++r) t += s[r] * ws[WS_W2 + tid * RK + r];
    ws[WS_COEF + tid] = 1.0f - 1.41421356237309515f * t;
  }
}

// ---------------------------------------------------------------------------
// Main fused kernel: per (batch, 64-channel tile)
//   H = Haar(x) in LDS (+ column sums -> mean), then
//   Y = W1^T * H  (WMMA f32 16x16x4, K=360), then
//   O = W2^T * Y + bias2 + mean*coef   (WMMA, K=32), store [336 x tile].
// ---------------------------------------------------------------------------
__global__ void __launch_bounds__(256) fused_main(const float* __restrict__ x,
                                                  const float* __restrict__ ws,
                                                  float* __restrict__ out) {
  extern __shared__ float sm[];
  float* sH    = sm + L_H;
  float* sY    = sm + L_Y;
  float* sW2   = sm + L_W2;
  float* sW1   = sm + L_W1;
  float* sBias = sm + L_BIAS;
  float* sCoef = sm + L_COEF;
  float* sCol  = sm + L_COL;

  const int tid = threadIdx.x;
  const int b   = blockIdx.y;
  const int c0  = blockIdx.x * NT;

  // ---- stage fused weights into LDS ----
  for (int e = tid; e < 32 * INL; e += 256) {
    const int r = e / INL, n = e - r * INL;
    sW1[r * W1P + n] = ws[WS_W1 + e];
  }
  for (int e = tid; e < PRED * RK; e += 256) {
    const int j = e >> 5, r = e & 31;
    sW2[j * W2P + r] = ws[WS_W2 + e];
  }
  for (int e = tid; e < PRED; e += 256) {
    sBias[e] = ws[WS_BIAS + e];
    sCoef[e] = ws[WS_COEF + e];
  }
  if (tid < NT) sCol[tid] = 0.f;
  __syncthreads();

  // ---- stage H (Haar low-pass of x), K-pair interleaved, + column sums ----
  {
    const int c  = tid & (NT - 1);     // channel within tile
    const int rg = tid >> 6;           // 0..3 row group
    const int gc = c0 + c;
    const bool valid = gc < CH;
    const float* xb = x + (size_t)b * SEQ * CH + gc;
    float acc = 0.f;
    for (int r = rg; r < INL; r += 4) {
      const float* p = xb + (size_t)(2 * r) * CH;
      __builtin_prefetch(p + 8 * (size_t)CH, 0, 1);
      float xa = valid ? p[0]  : 0.f;
      float xc = valid ? p[CH] : 0.f;
      float h  = (xa + xc) * 0.70710678118654752f;
      sH[(r >> 1) * HP2 + 2 * c + (r & 1)] = h;
      acc += h;
    }
    atomicAdd(&sCol[c], acc);          // ds_add_f32
  }
  __syncthreads();

  const int wave = tid >> 5;
  const int lane = tid & 31;
  const int half = lane >> 4;          // K-half select per ISA A/B layout
  const int lm   = lane & 15;

  // ---- GEMM1: Y[32 x NT] = W1^T[32x360] * H[360xNT] ----
  {
    const int mt = wave >> 2;          // 0..1
    const int nt = wave & 3;           // 0..3
    const int ncol = nt * 16 + lm;
    const float* pA = sW1 + (mt * 16 + lm) * W1P + 2 * half;  // +4 per step
    const float* pB = sH + half * HP2 + 2 * ncol;             // +2*HP2 per step
    v8f acc = {};
#pragma unroll 10
    for (int k = 0; k < INL; k += 4) {
      v2f a  = *reinterpret_cast<const v2f*>(pA);
      v2f bb = *reinterpret_cast<const v2f*>(pB);
      acc = __builtin_amdgcn_wmma_f32_16x16x4_f32(false, a, false, bb,
                                                  (short)0, acc, false, false);
      pA += 4;
      pB += 2 * HP2;
    }
#pragma unroll
    for (int i = 0; i < 8; ++i) {
      const int row = mt * 16 + i + 8 * half;
      sY[(row >> 1) * YP2 + 2 * ncol + (row & 1)] = acc[i];
    }
  }
  __syncthreads();

  // ---- GEMM2: O[336 x NT] = W2^T[336x32] * Y[32xNT], fused epilogue ----
  float* outb = out + (size_t)b * PRED * CH;
  for (int t = wave; t < 21 * 4; t += 8) {
    const int mt = t >> 2;             // 0..20
    const int nt = t & 3;
    const int ncol = nt * 16 + lm;
    const float* pA = sW2 + (mt * 16 + lm) * W2P + 2 * half;
    const float* pB = sY + half * YP2 + 2 * ncol;
    v8f acc = {};
#pragma unroll
    for (int k = 0; k < RK; k += 4) {
      v2f a  = *reinterpret_cast<const v2f*>(pA);
      v2f bb = *reinterpret_cast<const v2f*>(pB);
      acc = __builtin_amdgcn_wmma_f32_16x16x4_f32(false, a, false, bb,
                                                  (short)0, acc, false, false);
      pA += 4;
      pB += 2 * YP2;
    }
    const int gc = c0 + ncol;
    if (gc < CH) {
      const float mean = sCol[ncol] * (1.41421356237309515f / (float)SEQ);
#pragma unroll
      for (int i = 0; i < 8; ++i) {
        const int j = mt * 16 + i + 8 * half;
        outb[(size_t)j * CH + gc] = acc[i] + sBias[j] + mean * sCoef[j];
      }
    }
  }
}

// ---------------------------------------------------------------------------
extern "C" void kernel_launch(void* const* d_in, const int* in_sizes, int n_in,
                              void* d_out, int out_size, void* d_ws, size_t ws_size,
                              hipStream_t stream) {
  const float* x  = (const float*)d_in[0];
  const float* A  = (const float*)d_in[1];
  const float* Bm = (const float*)d_in[2];
  const float* bv = (const float*)d_in[3];
  float* ws = (float*)d_ws;
  float* o  = (float*)d_out;

  const int prep_threads = 32 * INL + PRED * RK + PRED;   // 22608
  prep_a<<<(prep_threads + 255) / 256, 256, 0, stream>>>(A, Bm, bv, ws);
  prep_b<<<1, 384, 0, stream>>>(ws);

  const size_t smem = (size_t)SMEM_DW * sizeof(float);    // ~220 KB dynamic LDS
  dim3 grid((CH + NT - 1) / NT, NB);
  fused_main<<<grid, 256, smem, stream>>>(x, ws, o);
}